// RelationalKENN_13271448944865
// MI455X (gfx1250) — compile-verified
//
#include <hip/hip_runtime.h>
#include <cstdint>
#include <cstddef>

// ---------------------------------------------------------------------------
// RelationalKENN for MI455X (gfx1250).
//
// Memory-bound scatter/gather workload (~90 MB HBM traffic -> ~4-8us roofline
// at 23.3 TB/s; FLOPs trivial). No GEMM exists in this op, so WMMA does not
// apply; the CDNA5-specific data-movement path used is the Tensor Data Mover
// (tensor_load_to_lds + s_wait_tensorcnt) staging the regular `binary`
// (2M x 4 f32) stream into LDS per workgroup, overlapped with the random
// gathers, plus non-temporal b128 stores for the streamed `bp` output so L2
// stays dedicated to the hot 1.6MB gather table and 6.4MB atomic region.
// ---------------------------------------------------------------------------

typedef float        f4  __attribute__((ext_vector_type(4)));
typedef unsigned int v4u __attribute__((ext_vector_type(4)));
typedef int          v8i __attribute__((ext_vector_type(8)));
typedef int          v4i __attribute__((ext_vector_type(4)));

#define NPRED 16
#define TPB   256          // threads per block (8 wave32 waves)
#define TILE  1024         // edges per workgroup (16 KB LDS tile)
#define EPT   (TILE / TPB) // edges per thread

__device__ __forceinline__ float fast_exp(float x) {
  // exp(x) = exp2(x * log2(e)) -> v_exp_f32 (transcendental, co-executes with VALU)
  return __builtin_exp2f(x * 1.44269504088896340736f);
}
__device__ __forceinline__ float fast_rcp(float x) { return __builtin_amdgcn_rcpf(x); }

// clause with literals (a:-1, b:+1): delta += w * sgn * softmax(sgn*x)
__device__ __forceinline__ void clause2(const float* x, float* d, int a, int b, float w) {
  float s0 = -x[a], s1 = x[b];
  float m  = fmaxf(s0, s1);
  float e0 = fast_exp(s0 - m), e1 = fast_exp(s1 - m);
  float inv = fast_rcp(e0 + e1);
  d[a] -= w * e0 * inv;
  d[b] += w * e1 * inv;
}
// clause with literals (a:-1, b:+1, c:+1)
__device__ __forceinline__ void clause3(const float* x, float* d, int a, int b, int c, float w) {
  float s0 = -x[a], s1 = x[b], s2 = x[c];
  float m  = fmaxf(s0, fmaxf(s1, s2));
  float e0 = fast_exp(s0 - m), e1 = fast_exp(s1 - m), e2 = fast_exp(s2 - m);
  float inv = fast_rcp(e0 + e1 + e2);
  d[a] -= w * e0 * inv;
  d[b] += w * e1 * inv;
  d[c] += w * e2 * inv;
}

// ---------------------------------------------------------------------------
// Kernel 1: u = unary + unary-clause enhancement.
// Writes full u into the `up` output region (base for later atomic scatter)
// and the compacted columns 0..3 into `uc` (gather table for the edge stage).
// ---------------------------------------------------------------------------
__global__ void __launch_bounds__(TPB)
kenn_node(const float* __restrict__ unary, const float* __restrict__ uw,
          float* __restrict__ up, f4* __restrict__ uc, int n_nodes) {
  int node = blockIdx.x * TPB + threadIdx.x;
  if (node >= n_nodes) return;

  const f4* row = (const f4*)(unary + (size_t)node * NPRED);
  f4 r0 = row[0], r1 = row[1], r2 = row[2], r3 = row[3];   // 4x global_load_b128

  float x[NPRED];
  #pragma unroll
  for (int j = 0; j < 4; ++j) { x[j] = r0[j]; x[4+j] = r1[j]; x[8+j] = r2[j]; x[12+j] = r3[j]; }

  float d[NPRED];
  #pragma unroll
  for (int j = 0; j < NPRED; ++j) d[j] = 0.0f;

  // UNARY_CLAUSES (hardcoded from the reference)
  clause2(x, d, 0, 1,      uw[0]);
  clause2(x, d, 1, 2,      uw[1]);
  clause3(x, d, 2, 3, 4,   uw[2]);
  clause2(x, d, 4, 5,      uw[3]);
  clause3(x, d, 6, 7, 8,   uw[4]);
  clause2(x, d, 8, 9,      uw[5]);
  clause3(x, d, 10, 11, 12, uw[6]);
  clause3(x, d, 13, 14, 15, uw[7]);

  f4 o0, o1, o2, o3;
  #pragma unroll
  for (int j = 0; j < 4; ++j) {
    o0[j] = x[j]      + d[j];
    o1[j] = x[4 + j]  + d[4 + j];
    o2[j] = x[8 + j]  + d[8 + j];
    o3[j] = x[12 + j] + d[12 + j];
  }
  f4* orow = (f4*)(up + (size_t)node * NPRED);
  orow[0] = o0; orow[1] = o1; orow[2] = o2; orow[3] = o3;  // 4x global_store_b128
  uc[node] = o0;                                           // compact gather table (cols 0..3)
}

// ---------------------------------------------------------------------------
// Kernel 2: edge stage. Binary tile staged into LDS via the Tensor Data Mover
// (async DMA, TENSORcnt); index loads + random gathers issued in the DMA
// shadow; non-temporal streamed bp output; hardware f32 atomics for the
// node scatter.
// ---------------------------------------------------------------------------
__global__ void __launch_bounds__(TPB)
kenn_edge(const float* __restrict__ binary, const int* __restrict__ idx1,
          const int* __restrict__ idx2, const float* __restrict__ bw,
          const f4* __restrict__ uc, float* __restrict__ up,
          float* __restrict__ bp, int n_edges) {
  __shared__ f4 sbin[TILE];   // 16 KB tile

  const int blockStart = blockIdx.x * TILE;

  // LDS byte offset of sbin: the generic (flat) LDS address keeps the LDS
  // offset in its low 32 bits (ISA 10.2 aperture mapping). The ptrtoint also
  // escapes sbin so the TDM's opaque LDS write can't be "proven dead".
  const unsigned int lds_base = (unsigned int)(uintptr_t)&sbin[0];

#if __has_builtin(__builtin_amdgcn_tensor_load_to_lds) && __has_builtin(__builtin_amdgcn_s_wait_tensorcnt)
  if (threadIdx.x < 32) {   // wave 0 issues the DMA once (TDM ignores EXEC; per-wave op)
    unsigned long long gaddr =
        (unsigned long long)(uintptr_t)(binary + (size_t)blockStart * 4);
    // Remaining f32 elements: tail-block reads past tensor_dim0 return zero (ISA 8.2).
    unsigned int rem = (unsigned int)(n_edges - blockStart) * 4u;

    // D# group 0: count=1, lds_addr=sbin, 57-bit global addr, type=2 ("image")
    v4u g0;
    g0.x = 1u;
    g0.y = lds_base;
    g0.z = (unsigned int)(gaddr & 0xffffffffull);
    g0.w = (unsigned int)((gaddr >> 32) & 0x01ffffffull) | (2u << 30);

    // D# group 1: data_size=4B (code 2), 1-D tile: tile_dim0 = TILE*4 elems,
    // tensor_dim0 = rem (OOB zero-fill), tensor_dim1 = 1, no pad/iterate/multicast.
    v8i g1;
    g1[0] = (int)(2u << 16);                                   // data_size=4B; wg_mask=0
    g1[1] = (int)((rem & 0xffffu) << 16);                      // tensor_dim0[15:0]
    g1[2] = (int)(((rem >> 16) & 0xffffu) | (1u << 16));       // tensor_dim0[31:16], tensor_dim1=1
    g1[3] = (int)((unsigned)(TILE * 4) << 16);                 // tile_dim0 = 4096 elems
    g1[4] = 0;                                                 // tile_dim1/2 unused (1-D)
    g1[5] = (int)rem;                                          // dim0 stride (unused for 1-D)
    g1[6] = 0;
    g1[7] = 0;

    v4i z4 = {0, 0, 0, 0};                  // groups 2/3 unused (<=2D tensor)
    v8i z8 = {0, 0, 0, 0, 0, 0, 0, 0};      // extra group in 6-arg builtin: unused
    __builtin_amdgcn_tensor_load_to_lds(g0, g1, z4, z4, z8, 0); // tensor_load_to_lds
    __builtin_amdgcn_s_wait_tensorcnt(0);                       // s_wait_tensorcnt 0
  }
#else
  // Fallback: cooperative coalesced copy.
  for (int i = threadIdx.x; i < TILE; i += TPB) {
    int e = blockStart + i;
    f4 z = {0.f, 0.f, 0.f, 0.f};
    sbin[i] = (e < n_edges) ? ((const f4*)binary)[e] : z;
  }
#endif

  // ---- Work that does NOT depend on the LDS tile: issue it in the DMA
  // shadow (index loads are coalesced b32; gathers hit the 1.6MB L2 table).
  int i1[EPT], i2[EPT];
  f4  u1[EPT], u2[EPT];
  #pragma unroll
  for (int k = 0; k < EPT; ++k) {
    int e = blockStart + threadIdx.x + k * TPB;
    bool valid = (e < n_edges);
    int a = valid ? idx1[e] : 0;   // node 0 is a safe dummy for tail lanes
    int b = valid ? idx2[e] : 0;
    i1[k] = a; i2[k] = b;
    u1[k] = uc[a];
    u2[k] = uc[b];
  }

  float wv0 = bw[0], wv1 = bw[1], wv2 = bw[2], wv3 = bw[3];
  float wv[4] = {wv0, wv1, wv2, wv3};

  // Keep the ds loads below ordered after the TDM write + barrier.
  asm volatile("" :: "v"(lds_base) : "memory");
  __syncthreads();

  #pragma unroll
  for (int k = 0; k < EPT; ++k) {
    int li = threadIdx.x + k * TPB;
    int e  = blockStart + li;
    if (e < n_edges) {
      f4 b4 = sbin[li];                 // ds_load_b128 from the TDM-staged tile

      f4 outb;
      float a1[4], a2[4];
      #pragma unroll
      for (int c = 0; c < 4; ++c) {
        // joined cols [c, 32+c, 16+c] with signs [-1, -1, +1]
        float s0 = -u1[k][c], s1 = -b4[c], s2 = u2[k][c];
        float m  = fmaxf(s0, fmaxf(s1, s2));
        float e0 = fast_exp(s0 - m), e1 = fast_exp(s1 - m), e2 = fast_exp(s2 - m);
        float inv = fast_rcp(e0 + e1 + e2);
        float w   = wv[c];
        outb[c] = b4[c] - w * e1 * inv;   // bp = binary + db
        a1[c]   = -w * e0 * inv;          // du1 scatter value
        a2[c]   =  w * e2 * inv;          // du2 scatter value
      }
      // Streamed, never re-read: non-temporal b128 store keeps L2 for hot data.
      __builtin_nontemporal_store(outb, (f4*)(bp + (size_t)e * 4));

      float* p1 = up + (size_t)i1[k] * NPRED;
      float* p2 = up + (size_t)i2[k] * NPRED;
      #pragma unroll
      for (int c = 0; c < 4; ++c) {
        unsafeAtomicAdd(p1 + c, a1[c]);   // global_atomic_add_f32 (no return)
        unsafeAtomicAdd(p2 + c, a2[c]);
      }
    }
  }
}

// ---------------------------------------------------------------------------
// Host launcher. Inputs (setup_inputs order):
//   0: unary (n_nodes*16 f32)   1: binary (n_edges*4 f32)
//   2: index1 (n_edges i32)     3: index2 (n_edges i32)
//   4: unary_weights (8 f32)    5: binary_weights (4 f32)
// Output: up (n_nodes*16 f32) ++ bp (n_edges*4 f32)
// Scratch: compact gather table uc (n_nodes * 4 f32 = 1.6 MB)
// ---------------------------------------------------------------------------
extern "C" void kernel_launch(void* const* d_in, const int* in_sizes, int n_in,
                              void* d_out, int out_size, void* d_ws, size_t ws_size,
                              hipStream_t stream) {
  const float* unary  = (const float*)d_in[0];
  const float* binary = (const float*)d_in[1];
  const int*   index1 = (const int*)d_in[2];
  const int*   index2 = (const int*)d_in[3];
  const float* uw     = (const float*)d_in[4];
  const float* bw     = (const float*)d_in[5];

  const int n_nodes = in_sizes[0] / NPRED;
  const int n_edges = in_sizes[1] / 4;

  float* up = (float*)d_out;
  float* bp = up + (size_t)n_nodes * NPRED;
  f4*    uc = (f4*)d_ws;

  const int nblk_nodes = (n_nodes + TPB - 1) / TPB;
  kenn_node<<<nblk_nodes, TPB, 0, stream>>>(unary, uw, up, uc, n_nodes);

  const int nblk_edges = (n_edges + TILE - 1) / TILE;
  kenn_edge<<<nblk_edges, TPB, 0, stream>>>(binary, index1, index2, bw,
                                            (const f4*)uc, up, bp, n_edges);
}